// HiPPO_LegT_21251498181478
// MI455X (gfx1250) — compile-verified
//
#include <hip/hip_runtime.h>
#include <hip/hip_bf16.h>

// HiPPO-LegT scan on gfx1250 via chunked linear-scan + V_WMMA_F32_16X16X4_F32.
//   c_t = c_{t-1} @ A^T + B * f_t,  state N=256, batch Bt=128, L=2048, f32.
//
// Phase 0: AT = A^T ; M64 = AT^64 via 6 WMMA GEMM squarings; pack AT/M64 into
//          the WMMA B-operand layout (one ds_load_b64 per WMMA B operand).
// Phase 1: per chunk g (T=64 steps), local scan with zero init -> U_g.
// Phase 2: serial over 32 chunks: S_{g+1} = S_g @ M64 + U_g   (8 WGs).
// Phase 3: per chunk, scan from S_g, writing all outputs (256 MB).
//
// Packed A^T (272KB) + double-buffered state live in the 320KB LDS of a WGP,
// staged with global_load_async_to_lds_b128 (ASYNCcnt path).

typedef __attribute__((ext_vector_type(2))) float v2f;
typedef __attribute__((ext_vector_type(8))) float v8f;

#define NST   256          // state size
#define BT    128          // batch
#define LSEQ  2048         // sequence length
#define TCH   64           // chunk length
#define GCH   32           // number of chunks (LSEQ/TCH)
#define STP   260          // LDS pitch for state (conflict-free A-op b64 reads)
#define BROWS 128          // packed B rows: 64 K-quads x 2 (kb=0/1)
#define BPG   512          // packed B global row stride (floats) = 256 cols x 2
#define BPL   544          // packed B LDS row stride; 544%64==32 -> no conflicts

// D(16x16) = A(16x4) * B(4x16) + C, f32.
__device__ __forceinline__ v8f wmma_f32(v2f a, v2f b, v8f c) {
    return __builtin_amdgcn_wmma_f32_16x16x4_f32(
        false, a, false, b, (short)0, c, false, false);
}

// gfx1250 async global->LDS copy (tracked by ASYNCcnt, no VGPR round-trip).
__device__ __forceinline__ void async_b128(unsigned lds_off, const float* g) {
    asm volatile("global_load_async_to_lds_b128 %0, %1, off"
                 :: "v"(lds_off), "v"((unsigned long long)(size_t)g)
                 : "memory");
}
__device__ __forceinline__ void wait_async0() {
    asm volatile("s_wait_asynccnt 0x0" ::: "memory");
}

// ---------------------------------------------------------------- transpose
__global__ void transpose_k(const float* __restrict__ A, float* __restrict__ AT) {
    int k = blockIdx.x;      // column of A -> row of AT
    int n = threadIdx.x;
    AT[k * NST + n] = A[n * NST + k];
}

// Pack row-major M(k,n) into WMMA B-operand pairs:
//   Mp[(k/2 + (k&1 ? ... )) ] : row r = (k>>2)*2 + (k&1), half h = (k>>1)&1
//   Mp[r*BPG + n*2 + h] = M[k*NST + n]   ->  lane b64 = {row k0+kb, row k0+kb+2}
__global__ void pack_b_k(const float* __restrict__ M, float* __restrict__ Mp) {
    int k = blockIdx.x, n = threadIdx.x;
    int r = ((k >> 2) << 1) + (k & 1);
    int h = (k >> 1) & 1;
    Mp[r * BPG + n * 2 + h] = M[k * NST + n];
}

// ------------------------------------------------- 256x256x256 f32 WMMA GEMM
// One wave per 16x16 tile; dual accumulators break the D->C dependence chain.
__global__ __launch_bounds__(256) void gemm256_k(const float* __restrict__ Ain,
                                                 const float* __restrict__ Bin,
                                                 float* __restrict__ Cout) {
    const int wave = threadIdx.x >> 5, lane = threadIdx.x & 31;
    const int tile = blockIdx.x * 8 + wave;          // 0..255
    const int m0 = (tile >> 4) << 4, n0 = (tile & 15) << 4;
    const int lm = lane & 15;
    const int kq = (lane < 16) ? 0 : 2;
    const int kb = (lane < 16) ? 0 : 1;
    const int mrow = m0 + lm, ncol = n0 + lm;

    v8f c0 = {}, c1 = {};
    #pragma unroll 2
    for (int k0 = 0; k0 < NST; k0 += 8) {
        v2f a0 = *(const v2f*)&Ain[mrow * NST + k0 + kq];
        v2f b0; b0.x = Bin[(k0 + kb) * NST + ncol];
                b0.y = Bin[(k0 + kb + 2) * NST + ncol];
        c0 = wmma_f32(a0, b0, c0);
        v2f a1 = *(const v2f*)&Ain[mrow * NST + k0 + 4 + kq];
        v2f b1; b1.x = Bin[(k0 + 4 + kb) * NST + ncol];
                b1.y = Bin[(k0 + 4 + kb + 2) * NST + ncol];
        c1 = wmma_f32(a1, b1, c1);
    }
    #pragma unroll
    for (int r = 0; r < 8; ++r) {
        int m = m0 + r + ((lane < 16) ? 0 : 8);
        Cout[m * NST + ncol] = c0[r] + c1[r];
    }
}

// --------------------------------------------------------------- chunk scan
// grid = (8 batch-tiles, 32 chunks); block = 512 threads = 16 waves.
__global__ __launch_bounds__(512) void chunk_scan_k(
    const float* __restrict__ inp,     // (L, BT)
    const float* __restrict__ Bpk,     // packed A^T (BROWS x BPG)
    const float* __restrict__ Bv,      // (256,)
    const float* __restrict__ Sinit,   // (G, BT, N) or null
    float* __restrict__ finals,        // (G, BT, N) or null
    float* __restrict__ out)           // (L, BT, N) or null
{
    __shared__ float lds_B[BROWS * BPL];         // 278528 B
    __shared__ float lds_st[2][16 * STP];        //  33280 B
    __shared__ float lds_f[TCH * 16];            //   4096 B  (total 315904 B)

    const int tid = threadIdx.x;
    const int b0 = blockIdx.x * 16;              // batch tile
    const int g  = blockIdx.y;                   // chunk

    // Async-stage packed A^T into LDS (128 rows x 128 float4).
    for (int i = tid; i < BROWS * (BPG / 4); i += 512) {
        int r = i >> 7, e = (i & 127) << 2;
        async_b128((unsigned)(size_t)&lds_B[r * BPL + e], &Bpk[r * BPG + e]);
    }
    // Stage this chunk's inputs for our 16 batch rows.
    for (int i = tid; i < TCH * 16; i += 512) {
        int t = i >> 4, m = i & 15;
        lds_f[i] = inp[(g * TCH + t) * BT + b0 + m];
    }
    // Initial state.
    for (int i = tid; i < 16 * NST; i += 512) {
        int m = i >> 8, n = i & 255;
        lds_st[0][m * STP + n] =
            Sinit ? Sinit[(size_t)g * BT * NST + (b0 + m) * NST + n] : 0.0f;
    }
    wait_async0();
    __syncthreads();

    const int wave = tid >> 5, lane = tid & 31;
    const int n0 = wave << 4;
    const int lm = lane & 15;
    const int kq = (lane < 16) ? 0 : 2;
    const int kb = (lane < 16) ? 0 : 1;
    const int mo = (lane < 16) ? 0 : 8;
    const int bcol = (n0 + lm) * 2;              // packed-B column offset
    const float bn = Bv[n0 + lm];

    int cur = 0;
    for (int t = 0; t < TCH; ++t) {
        const float* st = lds_st[cur];
        v8f c0 = {}, c1 = {};
        #pragma unroll 2
        for (int k0 = 0; k0 < NST; k0 += 8) {
            int r0 = (k0 >> 1) + kb;             // quad k0/4 -> rows k0/2+kb
            v2f a0 = *(const v2f*)&st[lm * STP + k0 + kq];       // ds_load_b64
            v2f b0 = *(const v2f*)&lds_B[r0 * BPL + bcol];       // ds_load_b64
            c0 = wmma_f32(a0, b0, c0);
            v2f a1 = *(const v2f*)&st[lm * STP + k0 + 4 + kq];
            v2f b1 = *(const v2f*)&lds_B[(r0 + 2) * BPL + bcol];
            c1 = wmma_f32(a1, b1, c1);
        }
        float* stn = lds_st[cur ^ 1];
        #pragma unroll
        for (int r = 0; r < 8; ++r) {
            int m = r + mo;
            float val = (c0[r] + c1[r]) + bn * lds_f[t * 16 + m];
            stn[m * STP + n0 + lm] = val;
            if (out)
                out[((size_t)(g * TCH + t) * BT + b0 + m) * NST + n0 + lm] = val;
        }
        __syncthreads();
        cur ^= 1;
    }

    if (finals) {
        for (int i = tid; i < 16 * NST; i += 512) {
            int m = i >> 8, n = i & 255;
            finals[(size_t)g * BT * NST + (b0 + m) * NST + n] =
                lds_st[cur][m * STP + n];
        }
    }
}

// --------------------------------------------- phase 2: inter-chunk combine
// S_0 = 0 ; S_{g+1} = S_g @ M64 + U_g.  8 blocks, 16 waves, 32 serial steps.
__global__ __launch_bounds__(512) void chain_k(const float* __restrict__ Mpk,
                                               const float* __restrict__ U,
                                               float* __restrict__ S) {
    __shared__ float lds_B[BROWS * BPL];         // 278528 B
    __shared__ float lds_st[2][16 * STP];        //  33280 B

    const int tid = threadIdx.x;
    const int b0 = blockIdx.x * 16;

    for (int i = tid; i < BROWS * (BPG / 4); i += 512) {
        int r = i >> 7, e = (i & 127) << 2;
        async_b128((unsigned)(size_t)&lds_B[r * BPL + e], &Mpk[r * BPG + e]);
    }
    for (int i = tid; i < 16 * NST; i += 512) {
        int m = i >> 8, n = i & 255;
        lds_st[0][m * STP + n] = 0.0f;
        S[(size_t)(b0 + m) * NST + n] = 0.0f;    // S_0 = 0
    }
    wait_async0();
    __syncthreads();

    const int wave = tid >> 5, lane = tid & 31;
    const int n0 = wave << 4;
    const int lm = lane & 15;
    const int kq = (lane < 16) ? 0 : 2;
    const int kb = (lane < 16) ? 0 : 1;
    const int mo = (lane < 16) ? 0 : 8;
    const int bcol = (n0 + lm) * 2;

    int cur = 0;
    for (int g = 0; g < GCH; ++g) {
        v8f c0, c1 = {};
        #pragma unroll
        for (int r = 0; r < 8; ++r)              // C := U_g tile
            c0[r] = U[(size_t)g * BT * NST + (b0 + r + mo) * NST + n0 + lm];

        const float* st = lds_st[cur];
        #pragma unroll 2
        for (int k0 = 0; k0 < NST; k0 += 8) {
            int r0 = (k0 >> 1) + kb;
            v2f a0 = *(const v2f*)&st[lm * STP + k0 + kq];
            v2f b0 = *(const v2f*)&lds_B[r0 * BPL + bcol];
            c0 = wmma_f32(a0, b0, c0);
            v2f a1 = *(const v2f*)&st[lm * STP + k0 + 4 + kq];
            v2f b1 = *(const v2f*)&lds_B[(r0 + 2) * BPL + bcol];
            c1 = wmma_f32(a1, b1, c1);
        }
        float* stn = lds_st[cur ^ 1];
        #pragma unroll
        for (int r = 0; r < 8; ++r) {
            int m = r + mo;
            float val = c0[r] + c1[r];
            stn[m * STP + n0 + lm] = val;
            if (g + 1 < GCH)
                S[(size_t)(g + 1) * BT * NST + (b0 + m) * NST + n0 + lm] = val;
        }
        __syncthreads();
        cur ^= 1;
    }
}

extern "C" void kernel_launch(void* const* d_in, const int* in_sizes, int n_in,
                              void* d_out, int out_size, void* d_ws, size_t ws_size,
                              hipStream_t stream) {
    const float* inp = (const float*)d_in[0];    // (2048, 128)
    const float* A   = (const float*)d_in[1];    // (256, 256)
    const float* Bv  = (const float*)d_in[2];    // (256,)
    float* out = (float*)d_out;

    float* ws  = (float*)d_ws;
    float* AT  = ws;                             // 65536 f (normal layout)
    float* W0  = ws + 65536;                     // 65536 f
    float* W1  = ws + 131072;                    // 65536 f
    float* ATp = ws + 196608;                    // 65536 f (packed B layout)
    float* M64p= ws + 262144;                    // 65536 f (packed B layout)
    float* U   = ws + 327680;                    // G*BT*N = 1048576 f
    float* S   = U + (size_t)GCH * BT * NST;     // 1048576 f  (~9.3 MB total)

    transpose_k<<<NST, NST, 0, stream>>>(A, AT);

    // M64 = AT^64 by repeated squaring (ends in W1).
    gemm256_k<<<32, 256, 0, stream>>>(AT, AT, W0);   // AT^2
    gemm256_k<<<32, 256, 0, stream>>>(W0, W0, W1);   // AT^4
    gemm256_k<<<32, 256, 0, stream>>>(W1, W1, W0);   // AT^8
    gemm256_k<<<32, 256, 0, stream>>>(W0, W0, W1);   // AT^16
    gemm256_k<<<32, 256, 0, stream>>>(W1, W1, W0);   // AT^32
    gemm256_k<<<32, 256, 0, stream>>>(W0, W0, W1);   // AT^64

    // Pack AT and M64 into the WMMA B-operand pair layout.
    pack_b_k<<<NST, NST, 0, stream>>>(AT, ATp);
    pack_b_k<<<NST, NST, 0, stream>>>(W1, M64p);

    // Phase 1: local chunk scans -> U (no outputs).
    chunk_scan_k<<<dim3(8, GCH), 512, 0, stream>>>(inp, ATp, Bv, nullptr, U, nullptr);
    // Phase 2: chunk-boundary states S.
    chain_k<<<8, 512, 0, stream>>>(M64p, U, S);
    // Phase 3: exact scans from S_g, write all outputs.
    chunk_scan_k<<<dim3(8, GCH), 512, 0, stream>>>(inp, ATp, Bv, S, nullptr, out);
}